// TemplateModel_62362925138565
// MI455X (gfx1250) — compile-verified
//
#include <hip/hip_runtime.h>
#include <cstddef>

#define HSMM_B 64
#define HSMM_K 128
#define HSMM_T 256
#define HSMM_L 16
#define NEGV   (-1e30f)
#define APAD   132   // padded row stride (words) for sA -> conflict-free WMMA gathers

typedef float v2f __attribute__((ext_vector_type(2)));
typedef float v8f __attribute__((ext_vector_type(8)));

__device__ __forceinline__ float blk_max128(float v, float* sRed, int tid) {
#pragma unroll
  for (int off = 16; off >= 1; off >>= 1)
    v = fmaxf(v, __shfl_xor(v, off, 32));
  if ((tid & 31) == 0) sRed[tid >> 5] = v;
  __syncthreads();
  v = fmaxf(fmaxf(sRed[0], sRed[1]), fmaxf(sRed[2], sRed[3]));
  __syncthreads();
  return v;
}

__device__ __forceinline__ float blk_sum128(float v, float* sRed, int tid) {
#pragma unroll
  for (int off = 16; off >= 1; off >>= 1)
    v += __shfl_xor(v, off, 32);
  if ((tid & 31) == 0) sRed[tid >> 5] = v;
  __syncthreads();
  v = sRed[0] + sRed[1] + sRed[2] + sRed[3];
  __syncthreads();
  return v;
}

__global__ __launch_bounds__(HSMM_K, 1)
void hsmm_bwd_dp_kernel(const float* __restrict__ init_logits,   // [B,K]
                        const float* __restrict__ trans_logits,  // [B,K,K]
                        const float* __restrict__ len_logits,    // [B,L]
                        const float* __restrict__ emis_lps,      // [B,K,T,L]
                        const float* __restrict__ pad_mask,      // [B,T]
                        float* __restrict__ ws_lse)              // [B]
{
  __shared__ float sA[HSMM_K * APAD];   // exp(trans_lps[b]) = softmax rows, padded
  __shared__ float sW[16 * HSMM_K];     // circular window of beta vectors
  __shared__ float sP[HSMM_K];          // p = exp(beta_star - bmax)
  __shared__ float sBS[HSMM_K];         // beta_star at current t
  __shared__ float sLen[HSMM_L];        // len log-probs
  __shared__ float sRed[8];             // reduction scratch

  const int b    = blockIdx.x;
  const int tid  = threadIdx.x;        // 0..127
  const int lane = tid & 31;
  const int wave = tid >> 5;           // 0..3

  // ---- lengths[b] = sum(pad_mask[b,:]) ----
  float pv = pad_mask[b * HSMM_T + tid] + pad_mask[b * HSMM_T + HSMM_K + tid];
  float plen = blk_sum128(pv, sRed, tid);
  const int len_b = (int)(plen + 0.5f);

  // ---- len log-softmax (tiny: thread 0) ----
  if (tid == 0) {
    float lx[HSMM_L];
    float lm = -3.0e38f;
#pragma unroll
    for (int l = 0; l < HSMM_L; ++l) { lx[l] = len_logits[b * HSMM_L + l]; lm = fmaxf(lm, lx[l]); }
    float ls = 0.0f;
#pragma unroll
    for (int l = 0; l < HSMM_L; ++l) ls += __expf(lx[l] - lm);
    float lse = lm + __logf(ls);
#pragma unroll
    for (int l = 0; l < HSMM_L; ++l) sLen[l] = lx[l] - lse;
  }

  // ---- window init: w0[0] = (len==T ? 0 : NEG), w0[1..15] = NEG ----
  sW[0 * HSMM_K + tid] = (len_b == HSMM_T) ? 0.0f : NEGV;
#pragma unroll
  for (int l = 1; l < 16; ++l) sW[l * HSMM_K + tid] = NEGV;

  // ---- exp(trans_lps[b]) into LDS: one wave per row, softmax of 128 values ----
  for (int r = wave; r < HSMM_K; r += 4) {
    const float* row = trans_logits + ((size_t)b * HSMM_K + r) * HSMM_K;
    float x0 = row[lane], x1 = row[lane + 32], x2 = row[lane + 64], x3 = row[lane + 96];
    float m = fmaxf(fmaxf(x0, x1), fmaxf(x2, x3));
#pragma unroll
    for (int off = 16; off >= 1; off >>= 1) m = fmaxf(m, __shfl_xor(m, off, 32));
    float e0 = __expf(x0 - m), e1 = __expf(x1 - m), e2 = __expf(x2 - m), e3 = __expf(x3 - m);
    float s = e0 + e1 + e2 + e3;
#pragma unroll
    for (int off = 16; off >= 1; off >>= 1) s += __shfl_xor(s, off, 32);
    float inv = 1.0f / s;
    float* dst = &sA[r * APAD];
    dst[lane] = e0 * inv; dst[lane + 32] = e1 * inv;
    dst[lane + 64] = e2 * inv; dst[lane + 96] = e3 * inv;
  }
  __syncthreads();

  int head = 0;  // sW[(head+l)&15] holds beta[t+1+l]

  for (int t = HSMM_T - 1; t >= 0; --t) {
    // ================= Phase A: beta_star (thread tid handles state k=tid) ===========
    const float4* e4 =
        (const float4*)(emis_lps + (((size_t)b * HSMM_K + tid) * HSMM_T + t) * HSMM_L);
    if (t) __builtin_prefetch((const void*)(e4 - 4), 0, 3);  // next step's emis row
    float4 q0 = e4[0], q1 = e4[1], q2 = e4[2], q3 = e4[3];
    float ev[16] = {q0.x, q0.y, q0.z, q0.w, q1.x, q1.y, q1.z, q1.w,
                    q2.x, q2.y, q2.z, q2.w, q3.x, q3.y, q3.z, q3.w};
    float vv[16];
#pragma unroll
    for (int l = 0; l < 16; ++l)
      vv[l] = sLen[l] + ev[l] + sW[((head + l) & 15) * HSMM_K + tid];
    float m = vv[0];
#pragma unroll
    for (int l = 1; l < 16; ++l) m = fmaxf(m, vv[l]);
    float s = 0.0f;
#pragma unroll
    for (int l = 0; l < 16; ++l) s += __expf(vv[l] - m);
    float bs = m + __logf(s);
    sBS[tid] = bs;
    __syncthreads();

    if (t == 0) {
      // loss term: LSE_k(beta_star0 + init_lps) = LSE_k(bs + x) - LSE_k(x)
      float xi = init_logits[b * HSMM_K + tid];
      float m1 = blk_max128(xi, sRed, tid);
      float s1 = blk_sum128(__expf(xi - m1), sRed, tid);
      float v  = sBS[tid] + xi;
      float m2 = blk_max128(v, sRed, tid);
      float s2 = blk_sum128(__expf(v - m2), sRed, tid);
      if (tid == 0) ws_lse[b] = (m2 + __logf(s2)) - (m1 + __logf(s1));
      break;
    }

    // ================= Phase B: p = exp(beta_star - bmax) ============================
    float bmax = blk_max128(sBS[tid], sRed, tid);
    sP[tid] = __expf(sBS[tid] - bmax);
    __syncthreads();

    // ================= Phase C: beta_exp = A * p via V_WMMA_F32_16X16X4_F32 ==========
    const int newhead = (head + 15) & 15;
    {
      const int sel = lane >> 4;            // 0: K={0,1}, 1: K={2,3} of each 4-chunk
      const int mlo = lane & 15;
#pragma unroll
      for (int ti = 0; ti < 2; ++ti) {
        const int tile = wave * 2 + ti;     // 8 row-tiles of 16 across 4 waves
        const float* Arow = &sA[(tile * 16 + mlo) * APAD];
        v8f acc0 = {0.f, 0.f, 0.f, 0.f, 0.f, 0.f, 0.f, 0.f};
        v8f acc1 = {0.f, 0.f, 0.f, 0.f, 0.f, 0.f, 0.f, 0.f};
#pragma unroll
        for (int c = 0; c < 32; c += 2) {
          const int kb0 = c * 4 + 2 * sel;
          const int kb1 = kb0 + 4;
          v2f a0; a0.x = Arow[kb0]; a0.y = Arow[kb0 + 1];
          v2f b0; b0.x = sP[kb0];   b0.y = sP[kb0 + 1];      // p replicated over N
          acc0 = __builtin_amdgcn_wmma_f32_16x16x4_f32(false, a0, false, b0,
                                                       (short)0, acc0, false, false);
          v2f a1; a1.x = Arow[kb1]; a1.y = Arow[kb1 + 1];
          v2f b1; b1.x = sP[kb1];   b1.y = sP[kb1 + 1];
          acc1 = __builtin_amdgcn_wmma_f32_16x16x4_f32(false, a1, false, b1,
                                                       (short)0, acc1, false, false);
        }
        v8f acc = acc0 + acc1;
        if ((lane & 15) == 0) {             // lanes 0 (rows 0..7) and 16 (rows 8..15)
          float* wdst = &sW[newhead * HSMM_K + tile * 16 + sel * 8];
          wdst[0] = acc[0]; wdst[1] = acc[1]; wdst[2] = acc[2]; wdst[3] = acc[3];
          wdst[4] = acc[4]; wdst[5] = acc[5]; wdst[6] = acc[6]; wdst[7] = acc[7];
        }
      }
    }
    __syncthreads();
    // beta = bmax + log(beta_exp), with boundary overrides (parallel over k)
    {
      float be = sW[newhead * HSMM_K + tid];
      float beta = (t == len_b) ? 0.0f
                 : ((t > len_b) ? NEGV : (bmax + __logf(be)));
      sW[newhead * HSMM_K + tid] = beta;
    }
    head = newhead;
    __syncthreads();
  }
}

__global__ void hsmm_finalize_kernel(const float* __restrict__ ws_lse,
                                     float* __restrict__ out) {
  __shared__ float sr[2];
  const int tid = threadIdx.x;  // 64 threads, 2 waves
  float v = ws_lse[tid];
#pragma unroll
  for (int off = 16; off >= 1; off >>= 1) v += __shfl_xor(v, off, 32);
  if ((tid & 31) == 0) sr[tid >> 5] = v;
  __syncthreads();
  if (tid == 0) out[0] = -(sr[0] + sr[1]) / (float)HSMM_B;
}

extern "C" void kernel_launch(void* const* d_in, const int* in_sizes, int n_in,
                              void* d_out, int out_size, void* d_ws, size_t ws_size,
                              hipStream_t stream) {
  (void)in_sizes; (void)n_in; (void)out_size; (void)ws_size;
  const float* init_logits  = (const float*)d_in[0];
  const float* trans_logits = (const float*)d_in[1];
  const float* len_logits   = (const float*)d_in[2];
  const float* emis_lps     = (const float*)d_in[3];
  const float* pad_mask     = (const float*)d_in[4];
  float* ws_lse = (float*)d_ws;           // 64 floats
  float* out    = (float*)d_out;          // 1 float

  hsmm_bwd_dp_kernel<<<HSMM_B, HSMM_K, 0, stream>>>(
      init_logits, trans_logits, len_logits, emis_lps, pad_mask, ws_lse);
  hsmm_finalize_kernel<<<1, HSMM_B, 0, stream>>>(ws_lse, out);
}